// NodeBlock_88734024336033
// MI455X (gfx1250) — compile-verified
//
#include <hip/hip_runtime.h>

#define N_NODES 100000
#define N_EDGES 1600000
#define D_NODE  64
#define D_EDGE  64
#define D_IN    128
#define D_HID   256
#define D_OUT   64

// W matrices live in LDS pre-swizzled into B-operand order:
//   element (k, col) -> (k>>1)*STR + col*2 + (k&1)
// so {W[k][col], W[k+1][col]} is one aligned ds_load_b64.
// Pair-row strides are 2*N + 32 (== 32 mod 64 banks): half-wave 0 (pair-row p)
// uses banks {0..31}, half-wave 1 (pair-row p+1) uses banks {32..63} -> conflict-free.
#define W1P_STR 544   // 2*256 + 32; 64 pair-rows -> 34816 floats (same as 128*272)
#define W2P_STR 160   // 2*64  + 32; 128 pair-rows -> 20480 floats (same as 256*80)
#define IN_STR  132   // 132 % 64 == 4 -> A-operand b64 reads hit 64 distinct banks
#define H_STR   260   // 260 % 64 == 4

#define LDS_W1_OFF 0
#define LDS_W2_OFF ((D_IN / 2) * W1P_STR)                // 34816
#define LDS_IN_OFF (LDS_W2_OFF + (D_HID / 2) * W2P_STR)  // 55296
#define LDS_H_OFF  (LDS_IN_OFF + 4 * 16 * IN_STR)        // 63744
#define LDS_B1_OFF (LDS_H_OFF + 4 * 16 * H_STR)          // 80384
#define LDS_B2_OFF (LDS_B1_OFF + D_HID)                  // 80640
#define LDS_TOTALF (LDS_B2_OFF + D_OUT)                  // 80704 floats = 322816 B

typedef __attribute__((ext_vector_type(2))) float v2f;
typedef __attribute__((ext_vector_type(8))) float v8f;

// ---------------- Kernel 1: zero the aggregation buffer ----------------
__global__ void zero_agg_kernel(float4* __restrict__ agg4, int n4) {
    int i = blockIdx.x * blockDim.x + threadIdx.x;
    if (i < n4) agg4[i] = make_float4(0.f, 0.f, 0.f, 0.f);
}

// ---------------- Kernel 2: scatter-add edge features ----------------
// one thread per (edge, 4-feature group); float4 load + 4 global f32 atomics
__global__ void scatter_add_kernel(const float4* __restrict__ edge_attr4,
                                   const long long* __restrict__ edge_index,
                                   float* __restrict__ agg) {
    long long idx = (long long)blockIdx.x * blockDim.x + threadIdx.x;
    const long long total = (long long)N_EDGES * (D_EDGE / 4);
    if (idx >= total) return;
    int e = (int)(idx >> 4);      // D_EDGE/4 == 16 groups per edge
    int g = (int)(idx & 15);
    long long recv = edge_index[(long long)N_EDGES + e];   // edge_index[1][e]
    float4 v = edge_attr4[idx];
    float* dst = agg + recv * (long long)D_EDGE + g * 4;
    atomicAdd(dst + 0, v.x);
    atomicAdd(dst + 1, v.y);
    atomicAdd(dst + 2, v.z);
    atomicAdd(dst + 3, v.w);
}

// ---------------- Kernel 3: fused 2-layer MLP with f32 WMMA ----------------
// block = 128 threads = 4 waves; each wave computes a 16-node tile.
// N-tile loops unrolled 4-wide: one A-operand b64 load feeds 4 independent
// WMMA chains; B operands are single b64 loads from the pre-swizzled W layout.
__global__ void fused_mlp_kernel(const float* __restrict__ x,
                                 const float* __restrict__ agg,
                                 const float* __restrict__ W1,
                                 const float* __restrict__ b1,
                                 const float* __restrict__ W2,
                                 const float* __restrict__ b2,
                                 float* __restrict__ out) {
    extern __shared__ float lds[];
    float* sW1 = lds + LDS_W1_OFF;
    float* sW2 = lds + LDS_W2_OFF;
    float* sB1 = lds + LDS_B1_OFF;
    float* sB2 = lds + LDS_B2_OFF;

    const int tid  = threadIdx.x;
    const int wave = tid >> 5;
    const int lane = tid & 31;
    const int m    = lane & 15;   // row (A) / column (B,C,D) index
    const int kh   = lane >> 4;   // which K-half this lane holds

    // ---- stage W1, W2 (swizzled to B-operand order), b1, b2 into LDS ----
    for (int i4 = tid; i4 < (D_IN * D_HID) / 4; i4 += blockDim.x) {
        float4 v = ((const float4*)W1)[i4];
        int i = i4 << 2;
        int k = i >> 8;            // source row (K index)
        int c = i & 255;           // source col, multiple of 4
        float* p = &sW1[(k >> 1) * W1P_STR + (k & 1) + c * 2];
        p[0] = v.x; p[2] = v.y; p[4] = v.z; p[6] = v.w;
    }
    for (int i4 = tid; i4 < (D_HID * D_OUT) / 4; i4 += blockDim.x) {
        float4 v = ((const float4*)W2)[i4];
        int i = i4 << 2;
        int k = i >> 6;
        int c = i & 63;
        float* p = &sW2[(k >> 1) * W2P_STR + (k & 1) + c * 2];
        p[0] = v.x; p[2] = v.y; p[4] = v.z; p[6] = v.w;
    }
    for (int i = tid; i < D_HID; i += blockDim.x) sB1[i] = b1[i];
    for (int i = tid; i < D_OUT; i += blockDim.x) sB2[i] = b2[i];
    __syncthreads();

    const int tileBase = blockIdx.x * 64 + wave * 16;
    float* sIn = lds + LDS_IN_OFF + wave * (16 * IN_STR);
    float* sH  = lds + LDS_H_OFF  + wave * (16 * H_STR);

    // ---- stage this wave's 16 node-input rows [x | agg] into LDS (coalesced) ----
    for (int r = 0; r < 16; ++r) {
        int node = tileBase + r;
        if (node >= N_NODES) node = N_NODES - 1;   // clamp; stores are guarded later
        for (int c = lane; c < D_NODE; c += 32) {
            sIn[r * IN_STR + c]          = x  [(long long)node * D_NODE + c];
            sIn[r * IN_STR + D_NODE + c] = agg[(long long)node * D_EDGE + c];
        }
    }

    // ---- layer 1: h = relu([x|agg] @ W1 + b1), 16x256 tile per wave ----
    for (int j = 0; j < D_HID / 16; j += 4) {
        v8f acc0 = {}, acc1 = {}, acc2 = {}, acc3 = {};
        #pragma unroll 4
        for (int k0 = 0; k0 < D_IN; k0 += 4) {
            int k = k0 + 2 * kh;                       // even
            v2f a = *(const v2f*)&sIn[m * IN_STR + k]; // one ds_load_b64
            const float* wp = &sW1[(k >> 1) * W1P_STR + m * 2];
            v2f bb0 = *(const v2f*)(wp + (j + 0) * 32);
            v2f bb1 = *(const v2f*)(wp + (j + 1) * 32);
            v2f bb2 = *(const v2f*)(wp + (j + 2) * 32);
            v2f bb3 = *(const v2f*)(wp + (j + 3) * 32);
            acc0 = __builtin_amdgcn_wmma_f32_16x16x4_f32(false, a, false, bb0, (short)0, acc0, false, false);
            acc1 = __builtin_amdgcn_wmma_f32_16x16x4_f32(false, a, false, bb1, (short)0, acc1, false, false);
            acc2 = __builtin_amdgcn_wmma_f32_16x16x4_f32(false, a, false, bb2, (short)0, acc2, false, false);
            acc3 = __builtin_amdgcn_wmma_f32_16x16x4_f32(false, a, false, bb3, (short)0, acc3, false, false);
        }
        #pragma unroll
        for (int t = 0; t < 4; ++t) {
            v8f acc = (t == 0) ? acc0 : (t == 1) ? acc1 : (t == 2) ? acc2 : acc3;
            float bias = sB1[(j + t) * 16 + m];
            #pragma unroll
            for (int r = 0; r < 8; ++r) {
                float h = acc[r] + bias;
                h = h > 0.f ? h : 0.f;
                sH[(r + 8 * kh) * H_STR + (j + t) * 16 + m] = h;
            }
        }
    }

    // ---- layer 2: out = h @ W2 + b2, all 4 output tiles in one K-sweep ----
    {
        v8f acc0 = {}, acc1 = {}, acc2 = {}, acc3 = {};
        #pragma unroll 4
        for (int k0 = 0; k0 < D_HID; k0 += 4) {
            int k = k0 + 2 * kh;
            v2f a = *(const v2f*)&sH[m * H_STR + k];
            const float* wp = &sW2[(k >> 1) * W2P_STR + m * 2];
            v2f bb0 = *(const v2f*)(wp + 0);
            v2f bb1 = *(const v2f*)(wp + 32);
            v2f bb2 = *(const v2f*)(wp + 64);
            v2f bb3 = *(const v2f*)(wp + 96);
            acc0 = __builtin_amdgcn_wmma_f32_16x16x4_f32(false, a, false, bb0, (short)0, acc0, false, false);
            acc1 = __builtin_amdgcn_wmma_f32_16x16x4_f32(false, a, false, bb1, (short)0, acc1, false, false);
            acc2 = __builtin_amdgcn_wmma_f32_16x16x4_f32(false, a, false, bb2, (short)0, acc2, false, false);
            acc3 = __builtin_amdgcn_wmma_f32_16x16x4_f32(false, a, false, bb3, (short)0, acc3, false, false);
        }
        #pragma unroll
        for (int t = 0; t < 4; ++t) {
            v8f acc = (t == 0) ? acc0 : (t == 1) ? acc1 : (t == 2) ? acc2 : acc3;
            float bias = sB2[t * 16 + m];
            #pragma unroll
            for (int r = 0; r < 8; ++r) {
                int node = tileBase + r + 8 * kh;
                if (node < N_NODES)
                    out[(long long)node * D_OUT + t * 16 + m] = acc[r] + bias;
            }
        }
    }
}

extern "C" void kernel_launch(void* const* d_in, const int* in_sizes, int n_in,
                              void* d_out, int out_size, void* d_ws, size_t ws_size,
                              hipStream_t stream) {
    const float*     x          = (const float*)d_in[0];
    const float*     edge_attr  = (const float*)d_in[1];
    const long long* edge_index = (const long long*)d_in[2];
    // d_in[3] = pos (unused by the node-update output)
    const float*     W1         = (const float*)d_in[4];
    const float*     b1         = (const float*)d_in[5];
    const float*     W2         = (const float*)d_in[6];
    const float*     b2         = (const float*)d_in[7];
    float*           out        = (float*)d_out;
    float*           agg        = (float*)d_ws;   // N_NODES*D_EDGE f32 = 25.6 MB

    // 1) zero agg
    int n4 = (N_NODES * D_EDGE) / 4;
    zero_agg_kernel<<<(n4 + 255) / 256, 256, 0, stream>>>((float4*)agg, n4);

    // 2) scatter-add edges -> agg
    long long total = (long long)N_EDGES * (D_EDGE / 4);
    int sblocks = (int)((total + 255) / 256);
    scatter_add_kernel<<<sblocks, 256, 0, stream>>>(
        (const float4*)edge_attr, edge_index, agg);

    // 3) fused MLP with fp32 WMMA, weights+biases resident in LDS
    int nblocks = (N_NODES + 63) / 64;
    size_t lds_bytes = (size_t)LDS_TOTALF * sizeof(float);   // 322816 B < 320 KB cap
    fused_mlp_kernel<<<nblocks, 128, lds_bytes, stream>>>(
        x, agg, W1, b1, W2, b2, out);
}